// Dual_enhancement_81647328297336
// MI455X (gfx1250) — compile-verified
//
#include <hip/hip_runtime.h>

// ---- problem geometry (fixed by reference) --------------------------------
#define NB        16                 // batch
#define CCH       512                // channels (rows of the [C, HW] matrix)
#define HW        1024               // h*w (cols)
#define BATCH_EL  (CCH * HW)         // 524288 floats per batch element
#define TENS_EL   (NB * BATCH_EL)    // 8388608 floats per tensor
#define BATCH_V4  (BATCH_EL / 4)     // 131072 float4 per batch element
#define TENS_V4   (TENS_EL / 4)      // 2097152 float4 per tensor

#define RED_CHUNK_V4   2048          // float4 per reduction block (256 thr x 8)
#define RED_BLKS_PER_T (TENS_V4 / RED_CHUNK_V4)   // 1024 blocks per tensor
#define RED_BLKS_PER_B (BATCH_V4 / RED_CHUNK_V4)  // 64 blocks per batch elem

typedef float v2f __attribute__((ext_vector_type(2)));
typedef float v4f __attribute__((ext_vector_type(4)));
typedef float v8f __attribute__((ext_vector_type(8)));

// ws layout: [0 .. 3071]  partial sums  partials[t][batch][blk64]
//            [3072..3119] final sums    sums[t][batch]
#define WS_PARTIALS 0
#define WS_SUMS     (3 * NB * RED_BLKS_PER_B)   // 3072

// ---------------------------------------------------------------------------
// Pass 1: per-block partial sums. Regular-temporal 128b loads (prime L2).
// Intra-wave reduction on the matrix pipe: V_WMMA_F32_16X16X4_F32 with
// A vgpr0 = lane partial, vgpr1 = 0 and B = all-ones gives
// D[m][n] = s[m] + s[m+16].  Summing the 8 D registers leaves lanes 0-15
// holding sum(rows 0..7) and lanes 16-31 holding sum(rows 8..15); lane 0 and
// lane 16 of each wave write their (uniform-within-half) value to LDS — no
// cross-lane shuffle needed at all.
// ---------------------------------------------------------------------------
__global__ __launch_bounds__(256) void batch_partial_sum_kernel(
    const float* __restrict__ a, const float* __restrict__ b,
    const float* __restrict__ c, float* __restrict__ ws)
{
    const int t = blockIdx.y;
    const float* __restrict__ src = (t == 0) ? a : ((t == 1) ? b : c);
    const v4f* __restrict__ p = (const v4f*)src;

    const long base = (long)blockIdx.x * RED_CHUNK_V4;
    float s = 0.0f;
#pragma unroll
    for (int k = 0; k < 8; ++k) {
        v4f v = p[base + threadIdx.x + k * 256];
        s += (v.x + v.y) + (v.z + v.w);
    }

    // ---- wave32 reduction via WMMA (EXEC all ones here: uniform flow) ----
    v2f A; A[0] = s;    A[1] = 0.0f;     // A: 16x4, K1 slot zeroed
    v2f B; B[0] = 1.0f; B[1] = 1.0f;     // B: 4x16 all ones (layout-agnostic)
    v8f Cm = {};
    v8f D = __builtin_amdgcn_wmma_f32_16x16x4_f32(
        /*neg_a=*/false, A, /*neg_b=*/false, B,
        /*c_mod=*/(short)0, Cm, /*reuse_a=*/false, /*reuse_b=*/false);
    const float half_sum =
        ((D[0] + D[1]) + (D[2] + D[3])) + ((D[4] + D[5]) + (D[6] + D[7]));

    // 8 waves x 2 half-writers = 16 LDS slots; threadIdx.x>>4 indexes them.
    __shared__ float wsum[16];
    if ((threadIdx.x & 15) == 0) wsum[threadIdx.x >> 4] = half_sum;
    __syncthreads();

    if (threadIdx.x == 0) {
        float bs = 0.0f;
#pragma unroll
        for (int w = 0; w < 16; ++w) bs += wsum[w];
        // partials[t][batch][blk-in-batch]  (deterministic: plain store)
        const int batch  = blockIdx.x / RED_BLKS_PER_B;
        const int blk_in = blockIdx.x % RED_BLKS_PER_B;
        ws[WS_PARTIALS + (t * NB + batch) * RED_BLKS_PER_B + blk_in] = bs;
    }
}

// ---------------------------------------------------------------------------
// Pass 1b: fold 64 partials per (tensor, batch) in a fixed order. Tiny.
// ---------------------------------------------------------------------------
__global__ __launch_bounds__(64) void fold_sums_kernel(float* __restrict__ ws)
{
    const int j = threadIdx.x;          // 48 active: t = j/16, batch = j%16
    if (j < 3 * NB) {
        const float* __restrict__ part = ws + WS_PARTIALS + j * RED_BLKS_PER_B;
        float s = 0.0f;
        for (int i = 0; i < RED_BLKS_PER_B; ++i) s += part[i];
        ws[WS_SUMS + j] = s;
    }
}

// ---------------------------------------------------------------------------
// Pass 2: elementwise.  One block == one (batch, channel) row of 1024 floats
// (256 lanes x float4), so batch / row / rand-block bounds / mean are all
// block-uniform -> SGPR loads + SALU; only the column mask is per-lane.
// NT 128b loads (last use; data is in L2 from pass 1) and NT 128b stores
// (192 MiB streamed out, never re-read -> don't evict L2).
// ---------------------------------------------------------------------------
__global__ __launch_bounds__(256) void enhance_kernel(
    const float* __restrict__ a, const float* __restrict__ b,
    const float* __restrict__ c, const int* __restrict__ rand_nums,
    const float* __restrict__ ws, float* __restrict__ out)
{
    const int t = blockIdx.y;
    const float* __restrict__ src = (t == 0) ? a : ((t == 1) ? b : c);

    // blockIdx.x in [0, NB*CCH): one full HW row per block.
    const int batch = blockIdx.x >> 9;        // / CCH
    const int row   = blockIdx.x & (CCH - 1); // channel index

    // random_num -> block bounds (faithful to from_num_to_block incl. quirk)
    const int rn  = rand_nums[t * NB + batch];
    const int rb  = (rn < 2) ? 1 : (rn - 1);
    const int idx = rb - 1;
    const int ir  = idx >> 2, ic = idx & 3;
    const int r0  = ir * (CCH / 4);
    const int r1  = (ir == 3) ? 5 : (ir + 1) * (CCH / 4);   // quirk: r+1
    const int c0  = ic * (HW / 4);
    const int c1  = (ic == 3) ? 5 : (ic + 1) * (HW / 4);    // quirk: c+1
    const bool rowin = (row >= r0) & (row < r1);            // block-uniform

    const float mean = ws[WS_SUMS + t * NB + batch] * (1.0f / (float)BATCH_EL);

    const long v4idx = (long)blockIdx.x * 256 + threadIdx.x;
    const int  col   = threadIdx.x << 2;      // first of 4 consecutive cols

    const v4f* __restrict__ pin = (const v4f*)src;
    v4f x = __builtin_nontemporal_load(&pin[v4idx]);

    v4f res, add;
#pragma unroll
    for (int j = 0; j < 4; ++j) {
        const float xi = x[j];
        const int   cj = col + j;
        const bool inblk = rowin & (cj >= c0) & (cj < c1);
        // tmp0 = 1 if x>mean; 0 if x<=mean && x<1; clip(x,0,1)=1 otherwise
        const float ib = (inblk & ((xi > mean) | (xi >= 1.0f))) ? 1.0f : 0.0f;
        res[j] = xi - 0.9f * xi * ib;   // keep FP expression of reference
        add[j] = xi + xi * ib;
    }

    v4f* __restrict__ od = (v4f*)(out + (long)t * TENS_EL);        // dv_t
    v4f* __restrict__ oa = (v4f*)(out + (long)(t + 3) * TENS_EL);  // add_t
    __builtin_nontemporal_store(res, &od[v4idx]);
    __builtin_nontemporal_store(add, &oa[v4idx]);
}

// ---------------------------------------------------------------------------
extern "C" void kernel_launch(void* const* d_in, const int* in_sizes, int n_in,
                              void* d_out, int out_size, void* d_ws, size_t ws_size,
                              hipStream_t stream)
{
    const float* fv = (const float*)d_in[0];
    const float* f1 = (const float*)d_in[1];
    const float* f2 = (const float*)d_in[2];
    const int*   rn = (const int*)d_in[3];
    float*       out = (float*)d_out;
    float*       ws  = (float*)d_ws;    // needs (3072 + 48) floats = 12.2 KB

    // Pass 1: partial sums (also primes L2 with the 96 MiB of input)
    dim3 g1(RED_BLKS_PER_T, 3, 1);
    batch_partial_sum_kernel<<<g1, 256, 0, stream>>>(fv, f1, f2, ws);

    // Pass 1b: deterministic fold to per-(tensor,batch) sums
    fold_sums_kernel<<<1, 64, 0, stream>>>(ws);

    // Pass 2: elementwise, one HW row per block, NT streaming stores
    dim3 g2(NB * CCH, 3, 1);            // 8192 x 3 blocks, 1 float4/thread
    enhance_kernel<<<g2, 256, 0, stream>>>(fv, f1, f2, rn, ws, out);
}